// DynamicSpatialRefine_51496657879119
// MI455X (gfx1250) — compile-verified
//
#include <hip/hip_runtime.h>
#include <hip/hip_bf16.h>
#include <math.h>

// DynamicSpatialRefine for MI455X (gfx1250, wave32).
// 3-kernel path (if workspace fits): pool+h -> batched dyn-weight GEMM (WMMA)
//   -> per-instance pipeline (dwconv + 2 WMMA GEMMs).
// Fallback single-kernel path otherwise (fused dyn-weight matvec on VALU).

#define WSZ 5
#define PP  25           // spatial positions per instance
#define CC  256          // channels
#define CP  (CC * PP)    // 6400 dynamic-weight outputs per instance
#define CDK 64           // dynamic-kernel hidden
#define CSE 32           // SE hidden
#define CG  32           // channels per group (G=8)
#define LDW 260          // padded LDS row stride (floats): 260%64==4 -> bank rotation
#define MT  32           // M padded 25 -> 32 for 16x16 WMMA tiles
#define NTHREADS 256

typedef __attribute__((ext_vector_type(2))) float v2f;
typedef __attribute__((ext_vector_type(8))) float v8f;

__device__ __forceinline__ float gelu_exact(float x) {
    // torch nn.GELU default: 0.5*x*(1+erf(x/sqrt(2)))
    return 0.5f * x * (1.0f + erff(x * 0.70710678118654752f));
}

__device__ __forceinline__ void wait_async0() {
#if __has_builtin(__builtin_amdgcn_s_wait_asynccnt)
    __builtin_amdgcn_s_wait_asynccnt(0);
#else
    asm volatile("s_wait_asynccnt 0x0" ::: "memory");
#endif
}

// Async DMA of one 16-byte chunk global -> LDS (ASYNCcnt-tracked, ISA 15.18.3).
__device__ __forceinline__ void async_g2l_b128(uint32_t lds_byte_off, const float* g) {
    asm volatile("global_load_async_to_lds_b128 %0, %1, off"
                 :: "v"(lds_byte_off), "v"(g) : "memory");
}

__device__ __forceinline__ v8f wmma4(v2f a, v2f b, v8f c) {
    return __builtin_amdgcn_wmma_f32_16x16x4_f32(false, a, false, b, (short)0, c,
                                                 false, false);
}

// D[0..31][0..255] = A[0..31][0..255] @ W^T  (W row-major [out=256][in=256]).
// A in LDS ([MT][LDW], rows 25..31 zero), D to LDS ([MT][LDW]).
// 16 N-tiles over 8 waves; each wave does 2 N-tiles with 2x2 register blocking:
// per k-step 2 A-frags + 1 B-frag (shared across both M-tiles) feed 2 WMMAs.
__device__ __forceinline__ void matmul_wmma_f32(const float* __restrict__ A_lds,
                                                const float* __restrict__ W,
                                                float* __restrict__ D_lds,
                                                int tid)
{
    const int wave = tid >> 5;
    const int lane = tid & 31;
    const int col  = lane & 15;          // row-within-tile for A, col for B/D
    const int koff = (lane >> 4) << 1;   // lanes 0-15: K={0,1}; lanes 16-31: K={2,3}

    const float* Ap0 = A_lds + col * LDW + koff;      // M-tile 0 (rows 0..15)
    const float* Ap1 = Ap0 + 16 * LDW;                // M-tile 1 (rows 16..31)

    for (int nt = wave; nt < 16; nt += 8) {           // 2 N-tiles per wave
        const int n0 = nt << 4;
        const float* Bp = W + (n0 + col) * CC + koff; // B[k][n] = W[n*CC + k]
        v8f acc0 = {};
        v8f acc1 = {};
#pragma unroll 8
        for (int k = 0; k < CC; k += 4) {
            v2f b  = *(const v2f*)(Bp  + k);
            v2f a0 = *(const v2f*)(Ap0 + k);
            v2f a1 = *(const v2f*)(Ap1 + k);
            acc0 = wmma4(a0, b, acc0);                // B frag reused
            acc1 = wmma4(a1, b, acc1);
        }
        // D layout: VGPR r -> (M = m0 + r + 8*(lane>=16), N = n0 + (lane&15))
        const int drow = (lane >> 4) << 3;
        float* Dp0 = D_lds + drow * LDW + (n0 + col);
        float* Dp1 = Dp0 + 16 * LDW;
#pragma unroll
        for (int r = 0; r < 8; ++r) {
            Dp0[r * LDW] = acc0[r];
            Dp1[r * LDW] = acc1[r];
        }
    }
}

// ---------------- kernel 0: per-instance spatial pool + hidden h ----------------
__global__ __launch_bounds__(NTHREADS)
void dsr_pool_h(const float* __restrict__ x,
                const float* __restrict__ dk_w1, const float* __restrict__ dk_b1,
                float* __restrict__ Hws)           // [N][64]
{
    __shared__ float sP[CC];
    const int n   = blockIdx.x;
    const int tid = threadIdx.x;
    const float* xg = x + (size_t)n * (PP * CC);
    float s = 0.f;
#pragma unroll
    for (int p = 0; p < PP; ++p) s += xg[p * CC + tid];   // coalesced per p
    sP[tid] = s * (1.f / 25.f);
    __syncthreads();
    if (tid < CDK) {
        const float* wrow = dk_w1 + tid * CC;
        float acc = dk_b1[tid];
        for (int c = 0; c < CC; ++c) acc += wrow[c] * sP[c];
        Hws[(size_t)n * CDK + tid] = gelu_exact(acc);
    }
}

// ---------------- kernel 1: batched dynamic-weight GEMM (WMMA) ----------------
// Wall[n][m] = sum_j H[n][j] * dk_w2[m][j]  + dw_w[m] + dk_b2[m]
// Grid: (ceil(Mtiles/2)) x (CP/(16*8)); block = 8 waves, 1 N-tile per wave,
// 2 M-tiles per block (B-frags of dk_w2 shared across both accumulators).
__global__ __launch_bounds__(NTHREADS)
void dsr_dyn_gemm(const float* __restrict__ Hws, const float* __restrict__ dk_w2,
                  const float* __restrict__ dw_w, const float* __restrict__ dk_b2,
                  float* __restrict__ Wall, int Mtiles)
{
    __shared__ __attribute__((aligned(16))) float sA[32 * 68]; // 32 x 64 (+4 pad)

    const int mt0  = blockIdx.x << 1;
    const bool two = (mt0 + 1) < Mtiles;     // block-uniform tail guard
    const int m0   = mt0 << 4;
    const int tid  = threadIdx.x;
    const int wave = tid >> 5;
    const int lane = tid & 31;

    // stage up to 32 rows of H (each thread: 8 floats of one row)
    {
        const int r = tid >> 3;              // 0..31
        const int c = (tid & 7) << 3;        // 0,8,..,56
        if (r < 16 || two) {
            const float4* src = (const float4*)(Hws + (size_t)(m0 + r) * CDK + c);
            float4 h0 = src[0], h1 = src[1];
            float* d = sA + r * 68 + c;
            d[0] = h0.x; d[1] = h0.y; d[2] = h0.z; d[3] = h0.w;
            d[4] = h1.x; d[5] = h1.y; d[6] = h1.z; d[7] = h1.w;
        }
    }
    __syncthreads();

    const int col  = lane & 15;
    const int koff = (lane >> 4) << 1;
    const int n0   = (blockIdx.y * 8 + wave) << 4;      // output column tile
    const float* Ap0 = sA + col * 68 + koff;
    const float* Ap1 = Ap0 + 16 * 68;
    const float* Bp  = dk_w2 + (size_t)(n0 + col) * CDK + koff;
    v8f acc0 = {};
    v8f acc1 = {};
#pragma unroll
    for (int k = 0; k < CDK; k += 4) {
        v2f b  = *(const v2f*)(Bp  + k);
        v2f a0 = *(const v2f*)(Ap0 + k);
        v2f a1 = *(const v2f*)(Ap1 + k);
        acc0 = wmma4(a0, b, acc0);
        acc1 = wmma4(a1, b, acc1);
    }
    const int ncol  = n0 + col;
    const float bias = dw_w[ncol] + dk_b2[ncol];
    const int drow  = (lane >> 4) << 3;
    float* Dp0 = Wall + (size_t)(m0 + drow) * CP + ncol;
#pragma unroll
    for (int r = 0; r < 8; ++r) Dp0[(size_t)r * CP] = acc0[r] + bias;
    if (two) {
        float* Dp1 = Dp0 + (size_t)16 * CP;
#pragma unroll
        for (int r = 0; r < 8; ++r) Dp1[(size_t)r * CP] = acc1[r] + bias;
    }
}

// ---------------- kernel 2: main per-instance pipeline ----------------
template <bool PRECOMP>
__global__ __launch_bounds__(NTHREADS)
void dsr_main(const float* __restrict__ x,
              const float* __restrict__ dw_w,
              const float* __restrict__ dk_w1, const float* __restrict__ dk_b1,
              const float* __restrict__ dk_w2, const float* __restrict__ dk_b2,
              const float* __restrict__ pw_w,
              const float* __restrict__ gc_w,  const float* __restrict__ gc_b,
              const float* __restrict__ se_w1, const float* __restrict__ se_b1,
              const float* __restrict__ se_w2, const float* __restrict__ se_b2,
              const float* __restrict__ proj_w,const float* __restrict__ proj_b,
              const float* __restrict__ alpha_p,
              const float* __restrict__ Wall,   // [N][6400] if PRECOMP
              float* __restrict__ out)
{
    __shared__ __attribute__((aligned(16))) float sBuf0[MT * LDW]; // xin -> pw out -> proj out
    __shared__ __attribute__((aligned(16))) float sBuf1[MT * LDW]; // dw out -> group/SE out
    __shared__ float sPooled[CC];
    __shared__ float sH[CDK];
    __shared__ float sHid[CSE];

    const int n   = blockIdx.x;
    const int tid = threadIdx.x;
    const float* xin_g = x + (size_t)n * (PP * CC);

    // warm L2-resident weight matrices (speculative prefetch, no counters)
    if (tid < 64) {
        __builtin_prefetch(pw_w   + (size_t)tid * 1024, 0, 0);
        __builtin_prefetch(proj_w + (size_t)tid * 1024, 0, 0);
    }

    // ---- 1) async DMA x into LDS rows 0..24; zero pad rows of BUF1 ----
    {
        const int r4 = tid >> 6;               // 0..3
        const int c4 = (tid & 63) << 2;        // 0,4,..,252
        const uint32_t lds0 = (uint32_t)(uintptr_t)&sBuf0[0];
#pragma unroll
        for (int pb = 0; pb < 24; pb += 4) {
            const int p = pb + r4;
            async_g2l_b128(lds0 + (uint32_t)((p * LDW + c4) << 2),
                           xin_g + p * CC + c4);
        }
        if (tid < 64)                          // row 24 tail (r4==0 here)
            async_g2l_b128(lds0 + (uint32_t)((24 * LDW + c4) << 2),
                           xin_g + 24 * CC + c4);
        for (int idx = tid; idx < (MT - PP) * LDW; idx += NTHREADS)
            sBuf1[PP * LDW + idx] = 0.f;
        wait_async0();
    }
    __syncthreads();

    // ---- 2..4) per-instance depthwise weights in registers (thread = channel) ----
    float wk[PP];
    if constexpr (PRECOMP) {
        const float* wrow = Wall + (size_t)n * CP + tid * PP;
#pragma unroll
        for (int p = 0; p < PP; ++p) wk[p] = wrow[p];
    } else {
        {   // spatial mean pool
            float s = 0.f;
#pragma unroll
            for (int p = 0; p < PP; ++p) s += sBuf0[p * LDW + tid];
            sPooled[tid] = s * (1.f / 25.f);
        }
        __syncthreads();
        if (tid < CDK) {    // h = GELU(dk_w1 @ pooled + dk_b1)
            const float* wrow = dk_w1 + tid * CC;
            float acc = dk_b1[tid];
            for (int c = 0; c < CC; ++c) acc += wrow[c] * sPooled[c];
            sH[tid] = gelu_exact(acc);
        }
        __syncthreads();
        const float* w2base = dk_w2 + (size_t)(tid * PP) * CDK;
#pragma unroll 5
        for (int p = 0; p < PP; ++p) {
            float acc = dw_w[tid * PP + p] + dk_b2[tid * PP + p];
            const float* w2 = w2base + p * CDK;
#pragma unroll 8
            for (int j = 0; j < CDK; ++j) acc += w2[j] * sH[j];
            wk[p] = acc;
        }
    }

    // ---- 5) depthwise 5x5 conv (pad=2) -> BUF1 rows 0..24 ----
    {
        const int c = tid;
#pragma unroll
        for (int i = 0; i < WSZ; ++i) {
#pragma unroll
            for (int j = 0; j < WSZ; ++j) {
                float acc = 0.f;
#pragma unroll
                for (int di = 0; di < WSZ; ++di) {
                    const int r = i + di - 2;
                    if (r < 0 || r >= WSZ) continue;
#pragma unroll
                    for (int dj = 0; dj < WSZ; ++dj) {
                        const int s2 = j + dj - 2;
                        if (s2 < 0 || s2 >= WSZ) continue;
                        acc += sBuf0[(r * WSZ + s2) * LDW + c] * wk[di * WSZ + dj];
                    }
                }
                sBuf1[(i * WSZ + j) * LDW + c] = acc;
            }
        }
    }
    __syncthreads();

    // ---- 6) pw 1x1 conv: BUF0 = BUF1 @ pw_w^T  (f32 WMMA, 2x2 blocked) ----
    matmul_wmma_f32(sBuf1, pw_w, sBuf0, tid);
    __syncthreads();

    // ---- 7) group conv (1x1, G=8) + bias -> BUF1 rows 0..24 (pad rows stay 0) ----
    {
        const int o = tid;
        const int g = o >> 5;
        const float* gw = gc_w + (g * CG + (o & 31)) * CG;   // gc_w[g][o%32][i]
        float acc[PP];
        const float b = gc_b[o];
#pragma unroll
        for (int p = 0; p < PP; ++p) acc[p] = b;
#pragma unroll 4
        for (int i = 0; i < CG; ++i) {
            const float wv = gw[i];
#pragma unroll
            for (int p = 0; p < PP; ++p)
                acc[p] += sBuf0[p * LDW + g * CG + i] * wv;
        }
#pragma unroll
        for (int p = 0; p < PP; ++p) sBuf1[p * LDW + o] = acc[p];
    }
    __syncthreads();

    // ---- 8) Squeeze-Excite ----
    {
        float s = 0.f;
#pragma unroll
        for (int p = 0; p < PP; ++p) s += sBuf1[p * LDW + tid];
        sPooled[tid] = s * (1.f / 25.f);
    }
    __syncthreads();
    if (tid < CSE) {
        const float* wrow = se_w1 + tid * CC;
        float acc = se_b1[tid];
        for (int c = 0; c < CC; ++c) acc += wrow[c] * sPooled[c];
        sHid[tid] = gelu_exact(acc);
    }
    __syncthreads();
    {
        const float* wrow = se_w2 + tid * CSE;
        float acc = se_b2[tid];
#pragma unroll
        for (int j = 0; j < CSE; ++j) acc += wrow[j] * sHid[j];
        const float s = 1.f / (1.f + __expf(-acc));
#pragma unroll
        for (int p = 0; p < PP; ++p) sBuf1[p * LDW + tid] *= s;
    }
    __syncthreads();

    // ---- 9) proj 1x1 conv: BUF0 = BUF1 @ proj_w^T  (f32 WMMA, 2x2 blocked) ----
    matmul_wmma_f32(sBuf1, proj_w, sBuf0, tid);
    __syncthreads();

    // ---- 10) bias + GELU + residual, coalesced store ----
    {
        const float alpha = *alpha_p;
        const float pb = proj_b[tid];
        float* og = out + (size_t)n * (PP * CC);
#pragma unroll
        for (int p = 0; p < PP; ++p) {
            const float v = gelu_exact(sBuf0[p * LDW + tid] + pb);
            og[p * CC + tid] = xin_g[p * CC + tid] + alpha * v;
        }
    }
}

extern "C" void kernel_launch(void* const* d_in, const int* in_sizes, int n_in,
                              void* d_out, int out_size, void* d_ws, size_t ws_size,
                              hipStream_t stream) {
    const float* x      = (const float*)d_in[0];
    const float* dw_w   = (const float*)d_in[1];
    const float* dk_w1  = (const float*)d_in[2];
    const float* dk_b1  = (const float*)d_in[3];
    const float* dk_w2  = (const float*)d_in[4];
    const float* dk_b2  = (const float*)d_in[5];
    const float* pw_w   = (const float*)d_in[6];
    const float* gc_w   = (const float*)d_in[7];
    const float* gc_b   = (const float*)d_in[8];
    const float* se_w1  = (const float*)d_in[9];
    const float* se_b1  = (const float*)d_in[10];
    const float* se_w2  = (const float*)d_in[11];
    const float* se_b2  = (const float*)d_in[12];
    const float* proj_w = (const float*)d_in[13];
    const float* proj_b = (const float*)d_in[14];
    const float* alpha  = (const float*)d_in[15];
    float* out = (float*)d_out;

    const int N = in_sizes[0] / (PP * CC);   // B*K = 3600
    const size_t need = (size_t)N * CDK * sizeof(float)      // H
                      + (size_t)N * CP  * sizeof(float);     // Wall

    if (ws_size >= need && (N & 15) == 0) {
        float* Hws  = (float*)d_ws;
        float* Wall = Hws + (size_t)N * CDK;
        const int Mtiles = N / 16;
        dsr_pool_h<<<N, NTHREADS, 0, stream>>>(x, dk_w1, dk_b1, Hws);
        dsr_dyn_gemm<<<dim3((Mtiles + 1) / 2, CP / (16 * 8)), NTHREADS, 0, stream>>>(
            Hws, dk_w2, dw_w, dk_b2, Wall, Mtiles);
        dsr_main<true><<<N, NTHREADS, 0, stream>>>(
            x, dw_w, dk_w1, dk_b1, dk_w2, dk_b2, pw_w, gc_w, gc_b,
            se_w1, se_b1, se_w2, se_b2, proj_w, proj_b, alpha, Wall, out);
    } else {
        dsr_main<false><<<N, NTHREADS, 0, stream>>>(
            x, dw_w, dk_w1, dk_b1, dk_w2, dk_b2, pw_w, gc_w, gc_b,
            se_w1, se_b1, se_w2, se_b2, proj_w, proj_b, alpha, nullptr, out);
    }
}